// DynamicRangeCompressor_47614007443842
// MI455X (gfx1250) — compile-verified
//
#include <hip/hip_runtime.h>
#include <hip/hip_bf16.h>

// Problem constants (fixed by the reference)
#define BATCH   64
#define TLEN    524288
#define DFACT   16
#define NFRAMES (BATCH * (TLEN / DFACT))      // 2,097,152 total frames
#define FROW    (TLEN / DFACT)                // 32,768 frames per row

typedef __attribute__((ext_vector_type(2))) float v2f;
typedef __attribute__((ext_vector_type(8))) float v8f;

// 20*log10(x) = 6.02059991328 * log2(x)
#define DB_SCALE 6.02059991328f
// 10^(v/20) = 2^(v * log2(10)/20)
#define EXP_SCALE 0.166096404744f
#define EPS 1e-7f

// ---------------------------------------------------------------------------
// Kernel 1: per-frame dB mean via V_WMMA_F32_16X16X4_F32 + static gain computer
// One wave = one tile of 16 consecutive frames (256 contiguous samples).
// A-matrix (16x4 f32, per ISA layout): lanes 0-15 hold K={0,1}; lanes 16-31 K={2,3}.
// B-matrix = all 1/16 (layout-invariant). 4 accumulating WMMAs cover K=0..15.
// D layout: VGPR v, lanes 0-15 -> M=v, lanes 16-31 -> M=v+8.
// ---------------------------------------------------------------------------
__global__ __launch_bounds__(256) void
drc_frame_gain_wmma(const float* __restrict__ audio,
                    const float* __restrict__ p_thr,
                    const float* __restrict__ p_ratio,
                    float* __restrict__ g)
{
    const float thr = p_thr[0];
    const float inv_ratio = 1.0f / p_ratio[0];

    const int lane      = threadIdx.x & 31;
    const int waveInBlk = threadIdx.x >> 5;
    const long tile     = (long)blockIdx.x * 8 + waveInBlk;   // 16 frames per tile
    const float* base   = audio + tile * 256;

    const int f  = lane & 15;   // frame within tile handled by this lane (A-matrix row M)
    const int hi = lane >> 4;   // 0: K-pair {0,1}; 1: K-pair {2,3}

    const v2f bmat = {0.0625f, 0.0625f};   // ones(4x16) / 16
    v8f c = {};

#pragma unroll
    for (int j = 0; j < 4; ++j) {
        // global K for this lane's two A elements: j*4 + hi*2 + {0,1}
        const float2 x = *(const float2*)(base + f * 16 + j * 4 + hi * 2);
        v2f a;
        a.x = DB_SCALE * __builtin_amdgcn_logf(__builtin_fabsf(x.x) + EPS);
        a.y = DB_SCALE * __builtin_amdgcn_logf(__builtin_fabsf(x.y) + EPS);
        c = __builtin_amdgcn_wmma_f32_16x16x4_f32(
                /*neg_a=*/false, a, /*neg_b=*/false, bmat,
                /*c_mod=*/(short)0, c, /*reuse_a=*/false, /*reuse_b=*/false);
    }

    // Lane 0 holds frame means 0..7 in c[0..7]; lane 16 holds frames 8..15.
    if ((lane & 15) == 0) {
        float* gout = g + tile * 16 + hi * 8;
#pragma unroll
        for (int v = 0; v < 8; ++v) {
            const float m = c[v];
            const float y = (m > thr) ? (thr + (m - thr) * inv_ratio) : m;
            gout[v] = y - m;               // desired gain in dB (<= 0)
        }
    }
}

// ---------------------------------------------------------------------------
// Kernel 2: exact attack/release one-pole IIR, one lane per batch row.
// Critical path per step: d = prev-g; alpha = (d>0) ? at : rt; prev = fma(alpha,d,g).
// Converts smoothed dB gain (+makeup) to a LINEAR multiplier per frame, so the
// per-sample kernel needs no transcendentals.
// ---------------------------------------------------------------------------
__global__ __launch_bounds__(32) void
drc_scan(const float* __restrict__ g,
         const float* __restrict__ p_makeup,
         const float* __restrict__ p_attack,
         const float* __restrict__ p_release,
         float* __restrict__ lin)
{
    const int row = blockIdx.x * 32 + threadIdx.x;    // 0..63
    const float at = p_attack[0];
    const float rt = p_release[0];
    const float mk = p_makeup[0];

    const float* gr = g   + (size_t)row * FROW;
    float*       lr = lin + (size_t)row * FROW;

    float prev = 0.0f;
    for (int i = 0; i < FROW; i += 4) {
        const float4 gt = *(const float4*)(gr + i);
        float4 o;
        float d, a;
        d = prev - gt.x; a = (d > 0.0f) ? at : rt; prev = __builtin_fmaf(a, d, gt.x);
        o.x = __builtin_amdgcn_exp2f((prev + mk) * EXP_SCALE);
        d = prev - gt.y; a = (d > 0.0f) ? at : rt; prev = __builtin_fmaf(a, d, gt.y);
        o.y = __builtin_amdgcn_exp2f((prev + mk) * EXP_SCALE);
        d = prev - gt.z; a = (d > 0.0f) ? at : rt; prev = __builtin_fmaf(a, d, gt.z);
        o.z = __builtin_amdgcn_exp2f((prev + mk) * EXP_SCALE);
        d = prev - gt.w; a = (d > 0.0f) ? at : rt; prev = __builtin_fmaf(a, d, gt.w);
        o.w = __builtin_amdgcn_exp2f((prev + mk) * EXP_SCALE);
        *(float4*)(lr + i) = o;
    }
}

// ---------------------------------------------------------------------------
// Kernel 3: apply linear gain, one thread per frame (16 samples, 4x float4).
// Pure streaming: ~270 MB @ 23.3 TB/s.
// ---------------------------------------------------------------------------
__global__ __launch_bounds__(256) void
drc_apply(const float* __restrict__ audio,
          const float* __restrict__ lin,
          float* __restrict__ out)
{
    const long fr = (long)blockIdx.x * 256 + threadIdx.x;     // frame index
    const float gl = lin[fr];
    const float4* a = (const float4*)(audio + fr * 16);
    float4*       o = (float4*)(out   + fr * 16);
#pragma unroll
    for (int j = 0; j < 4; ++j) {
        float4 x = a[j];
        x.x *= gl; x.y *= gl; x.z *= gl; x.w *= gl;
        o[j] = x;
    }
}

extern "C" void kernel_launch(void* const* d_in, const int* in_sizes, int n_in,
                              void* d_out, int out_size, void* d_ws, size_t ws_size,
                              hipStream_t stream)
{
    const float* audio     = (const float*)d_in[0];
    const float* threshold = (const float*)d_in[1];
    const float* ratio     = (const float*)d_in[2];
    const float* makeup    = (const float*)d_in[3];
    const float* attack    = (const float*)d_in[4];
    const float* release   = (const float*)d_in[5];
    float* out = (float*)d_out;

    // Scratch layout: g [NFRAMES] floats, lin [NFRAMES] floats (16 MB total)
    float* g   = (float*)d_ws;
    float* lin = g + NFRAMES;

    // 1) frame dB mean (WMMA reduction) + static gain computer
    //    NFRAMES/16 tiles, 8 waves (tiles) per 256-thread block
    const int tiles  = NFRAMES / 16;          // 131072
    const int blocks = tiles / 8;             // 16384
    drc_frame_gain_wmma<<<blocks, 256, 0, stream>>>(audio, threshold, ratio, g);

    // 2) exact serial envelope follower, 64 rows on 64 lanes
    drc_scan<<<2, 32, 0, stream>>>(g, makeup, attack, release, lin);

    // 3) apply linear gain at audio rate
    drc_apply<<<NFRAMES / 256, 256, 0, stream>>>(audio, lin, out);
}